// NODE_88880053223712
// MI455X (gfx1250) — compile-verified
//
#include <hip/hip_runtime.h>
#include <math.h>

// ---------------------------------------------------------------------------
// Tsit5 neural-ODE (SEIAR + MLP beta(t)) for MI455X / gfx1250.
//
// Strategy: beta(t) depends only on t, and all evaluation times are known up
// front. Phase 1 evaluates all (nts-1)*20*6 betas as a batched MLP where the
// 64x64 hidden layer is a fp32 WMMA GEMM (V_WMMA_F32_16X16X4_F32). Phase 2 is
// the unavoidable sequential 5-dim Tsit5 scan, reading precomputed betas
// (software-pipelined loads, all resident in L2).
// ---------------------------------------------------------------------------

typedef __attribute__((ext_vector_type(2))) float v2f;
typedef __attribute__((ext_vector_type(8))) float v8f;

#define N_SUB   20
#define N_STAGE 6
#define HID     64

__constant__ float c_stage[N_STAGE] = {
    0.0f, 0.161f, 0.327f, 0.9f, 0.9800255409045097f, 1.0f};

__device__ __forceinline__ float softplus_f(float x) {
    // jax.nn.softplus = log1p(exp(x)); stable for large x
    return (x > 20.0f) ? x : log1pf(expf(x));
}

// ---------------------------------------------------------------------------
// Phase 1: one wave (32 lanes) per tile of 16 time-points.
//   layer0: h1 = softplus(W0*t + b0)            -> LDS, k-major [64][16]
//   layer1: pre = W1 @ h1  via 4 Mtiles x 16 Ksteps of WMMA f32 16x16x4
//   layer2: o = W2 . softplus(pre + b1) ; beta = sigmoid(1e-4*(o+b2))
// ---------------------------------------------------------------------------
__global__ void __launch_bounds__(32)
beta_mlp_kernel(const float* __restrict__ ts,
                const float* __restrict__ W0, const float* __restrict__ b0,
                const float* __restrict__ W1, const float* __restrict__ b1,
                const float* __restrict__ W2, const float* __restrict__ b2,
                float* __restrict__ betas, int n_beta) {
    __shared__ float H1[HID * 16];          // k-major: H1[k*16 + n]

    const int lane  = threadIdx.x;          // 0..31 (wave32)
    const int n     = lane & 15;            // time slot within tile (N dim)
    const int khalf = lane >> 4;            // 0: K0/K1 half, 1: K2/K3 half
    const int g     = blockIdx.x * 16 + n;  // global beta slot

    // ---- evaluation time for this slot -----------------------------------
    float t;
    {
        int gg = (g < n_beta) ? g : 0;      // pad slots recompute slot 0
        int s  = gg % N_STAGE;
        int u  = gg / N_STAGE;
        int i  = u % N_SUB;
        int m  = u / N_SUB;                 // < nts-1 by construction
        float t0 = ts[m];
        float dt = (ts[m + 1] - t0) * (1.0f / (float)N_SUB);
        t = fmaf((float)i + c_stage[s], dt, t0);
    }

    // ---- layer 0 into LDS (each lane: 32 of the 64 hidden units) ---------
    #pragma unroll
    for (int j = 0; j < 32; ++j) {
        int k = khalf * 32 + j;
        H1[k * 16 + n] = softplus_f(fmaf(W0[k], t, b0[k]));
    }
    __syncthreads();

    // ---- layer 1 (WMMA) + layer 2 (per-lane partial dot) -----------------
    float partial = 0.0f;
    #pragma unroll
    for (int mt = 0; mt < 4; ++mt) {        // 4 tiles of 16 output rows
        v8f acc = {};
        const int rowA = mt * 16 + n;       // A-matrix row for this lane
        #pragma unroll
        for (int kt = 0; kt < 16; ++kt) {   // K = 64 in steps of 4
            const int k0 = kt * 4 + khalf * 2;
            // A frag (16x4 f32): lanes 0-15 hold K0,K1; lanes 16-31 K2,K3
            v2f a;
            a.x = W1[rowA * HID + k0];
            a.y = W1[rowA * HID + k0 + 1];
            // B frag (4x16 f32), conflict-free k-major LDS reads
            v2f bf;
            bf.x = H1[(k0    ) * 16 + n];
            bf.y = H1[(k0 + 1) * 16 + n];
            acc = __builtin_amdgcn_wmma_f32_16x16x4_f32(
                false, a, false, bf, (short)0, acc, false, false);
        }
        // C/D layout: VGPR v holds M = mt*16 + (khalf?8:0) + v, N = n
        #pragma unroll
        for (int v = 0; v < 8; ++v) {
            const int row = mt * 16 + khalf * 8 + v;
            float h2 = softplus_f(acc[v] + b1[row]);
            partial = fmaf(W2[row], h2, partial);
        }
    }

    // combine the two lane-halves (each covers complementary rows)
    partial += __shfl_xor(partial, 16, 32);

    if ((lane < 16) && (g < n_beta)) {
        float o = partial + b2[0];
        betas[g] = 1.0f / (1.0f + expf(-1e-4f * o));   // sigmoid
    }
}

// ---------------------------------------------------------------------------
// Phase 2: sequential Tsit5 scan, 5-dim state, betas precomputed.
// ---------------------------------------------------------------------------
__device__ __forceinline__ void rhs5(float bb, const float* y, float* k) {
    // kk=0.526, aa=0.244, ii=0.244, p=0.667, f=0.98, ee=0, dd=1, q=0.5
    const float kk = 0.526f, aa = 0.244f, ii = 0.244f;
    const float p = 0.667f, f = 0.98f, ee = 0.0f, dd = 1.0f, q = 0.5f;
    float LL  = ee * y[1] + (1.0f - q) * y[2] + dd * y[3];
    float inf = bb * y[0] * LL;
    k[0] = -inf;
    k[1] = inf - kk * y[1];
    k[2] = p * kk * y[1] - aa * y[2];
    k[3] = (1.0f - p) * kk * y[1] - ii * y[3];
    k[4] = f * aa * y[2] + ii * y[3];
}

__global__ void __launch_bounds__(32)
tsit5_scan_kernel(const float* __restrict__ y0, const float* __restrict__ ts,
                  const float* __restrict__ betas, float* __restrict__ out,
                  int n_intervals, int n_beta) {
    if (threadIdx.x != 0 || blockIdx.x != 0) return;

    const float A21 = 0.161f;
    const float A31 = -0.008480655492356989f, A32 = 0.335480655492357f;
    const float A41 = 2.8971530571054935f,  A42 = -6.359448489975075f,  A43 = 4.3622954328695815f;
    const float A51 = 5.325864828439257f,   A52 = -11.748883564062828f, A53 = 7.4955393428898365f, A54 = -0.09249506636175525f;
    const float A61 = 5.86145544294642f,    A62 = -12.92096931784711f,  A63 = 8.159367898576159f,  A64 = -0.071584973281401f, A65 = -0.028269050394068383f;
    const float B1 = 0.09646076681806523f, B2 = 0.01f, B3 = 0.4798896504144996f;
    const float B4 = 1.379008574103742f,   B5 = -3.290069515436081f, B6 = 2.324710524099774f;

    float y[5];
    #pragma unroll
    for (int c = 0; c < 5; ++c) { y[c] = y0[c]; out[c] = y[c]; }

    // software pipeline: bn holds the next substep's 6 betas
    float bn[6];
    #pragma unroll
    for (int s = 0; s < 6; ++s) bn[s] = betas[s];
    int nextb = 6;

    for (int m = 0; m < n_intervals; ++m) {
        const float t0 = ts[m];
        const float dt = (ts[m + 1] - t0) * (1.0f / (float)N_SUB);

        for (int i = 0; i < N_SUB; ++i) {
            float b[6];
            #pragma unroll
            for (int s = 0; s < 6; ++s) b[s] = bn[s];
            // issue next substep's loads now; latency hides under compute
            int nb = (nextb < n_beta) ? nextb : 0;
            #pragma unroll
            for (int s = 0; s < 6; ++s) bn[s] = betas[nb + s];
            nextb += 6;

            float k1[5], k2[5], k3[5], k4[5], k5[5], k6[5], yt[5];
            rhs5(b[0], y, k1);
            #pragma unroll
            for (int c = 0; c < 5; ++c) yt[c] = fmaf(dt * A21, k1[c], y[c]);
            rhs5(b[1], yt, k2);
            #pragma unroll
            for (int c = 0; c < 5; ++c) yt[c] = y[c] + dt * (A31 * k1[c] + A32 * k2[c]);
            rhs5(b[2], yt, k3);
            #pragma unroll
            for (int c = 0; c < 5; ++c) yt[c] = y[c] + dt * (A41 * k1[c] + A42 * k2[c] + A43 * k3[c]);
            rhs5(b[3], yt, k4);
            #pragma unroll
            for (int c = 0; c < 5; ++c) yt[c] = y[c] + dt * (A51 * k1[c] + A52 * k2[c] + A53 * k3[c] + A54 * k4[c]);
            rhs5(b[4], yt, k5);
            #pragma unroll
            for (int c = 0; c < 5; ++c) yt[c] = y[c] + dt * (A61 * k1[c] + A62 * k2[c] + A63 * k3[c] + A64 * k4[c] + A65 * k5[c]);
            rhs5(b[5], yt, k6);
            #pragma unroll
            for (int c = 0; c < 5; ++c)
                y[c] = y[c] + dt * (B1 * k1[c] + B2 * k2[c] + B3 * k3[c] +
                                    B4 * k4[c] + B5 * k5[c] + B6 * k6[c]);
        }
        #pragma unroll
        for (int c = 0; c < 5; ++c) out[(m + 1) * 5 + c] = y[c];
    }
}

// ---------------------------------------------------------------------------
extern "C" void kernel_launch(void* const* d_in, const int* in_sizes, int n_in,
                              void* d_out, int out_size, void* d_ws, size_t ws_size,
                              hipStream_t stream) {
    (void)n_in; (void)out_size; (void)ws_size;
    const float* y0 = (const float*)d_in[0];
    const float* ts = (const float*)d_in[1];
    const float* W0 = (const float*)d_in[2];
    const float* b0 = (const float*)d_in[3];
    const float* W1 = (const float*)d_in[4];
    const float* b1 = (const float*)d_in[5];
    const float* W2 = (const float*)d_in[6];
    const float* b2 = (const float*)d_in[7];
    float* out   = (float*)d_out;
    float* betas = (float*)d_ws;            // (nts-1)*120 floats of scratch

    const int nts         = in_sizes[1];
    const int n_intervals = nts - 1;
    const int n_beta      = n_intervals * N_SUB * N_STAGE;
    const int n_tiles     = (n_beta + 15) / 16;

    beta_mlp_kernel<<<n_tiles, 32, 0, stream>>>(ts, W0, b0, W1, b1, W2, b2,
                                                betas, n_beta);
    tsit5_scan_kernel<<<1, 32, 0, stream>>>(y0, ts, betas, out,
                                            n_intervals, n_beta);
}